// LSTMScratch_77283641524311
// MI455X (gfx1250) — compile-verified
//
#include <hip/hip_runtime.h>
#include <cstddef>

#define T_STEPS 1024
#define BATCH   64
#define DIN     512
#define HDIM    512

typedef __attribute__((ext_vector_type(2))) float v2f;
typedef __attribute__((ext_vector_type(8))) float v8f;

// D = A(16x4 f32) * B(4x16 f32) + C(16x16 f32)
__device__ __forceinline__ v8f wmma4(v2f a, v2f b, v8f c) {
  return __builtin_amdgcn_wmma_f32_16x16x4_f32(
      /*neg_a=*/false, a, /*neg_b=*/false, b,
      /*c_mod=*/(short)0, c, /*reuse_a=*/false, /*reuse_b=*/false);
}

// ---------------------------------------------------------------------------
// Packed weight layout: Wp[r][col][2] = { W[2r][col], W[2r+1][col] },
// r in [0, K/2), col in [0, 512). A lane's B fragment (K-pair at one column)
// becomes a single aligned 8-byte load.
// ---------------------------------------------------------------------------
struct PackParams { const float* src[8]; float* dst; };

__global__ __launch_bounds__(256)
void pack_weights(PackParams p) {
  // 8 matrices x 256 pair-rows x 512 cols = 1,048,576 pairs
  unsigned idx = blockIdx.x * 256u + threadIdx.x;
  unsigned m   = idx >> 17;          // matrix 0..7
  unsigned rem = idx & 131071u;      // r*512 + col
  unsigned r   = rem >> 9;
  unsigned col = rem & 511u;
  const float* W = p.src[m];
  v2f v;
  v.x = W[(size_t)(2 * r) * HDIM + col];
  v.y = W[(size_t)(2 * r + 1) * HDIM + col];
  *((v2f*)p.dst + idx) = v;
}

// ---------------------------------------------------------------------------
// One wave computes a 32-row x 64-col strip of A[32 x K] * W[K x 512].
// A: row-major, base already offset to the 32-row block. Wp: packed pairs.
// acc[a][nt]: a = 16-row half, nt = 16-col subtile.
// ---------------------------------------------------------------------------
__device__ __forceinline__ void gemm_strip32(const float* __restrict__ A, int K,
                                             const float* __restrict__ Wp,
                                             int n0, int hf, int ln,
                                             v8f (&acc)[2][4]) {
  const float* Ar0 = A + (size_t)ln * DIN;        // DIN == HDIM == 512 stride
  const float* Ar1 = Ar0 + 16 * (size_t)DIN;
  const v2f* WpV = (const v2f*)Wp;
#pragma unroll 2
  for (int k0 = 0; k0 < K; k0 += 4) {
    v2f a0 = *(const v2f*)(Ar0 + k0 + hf * 2);
    v2f a1 = *(const v2f*)(Ar1 + k0 + hf * 2);
    const v2f* wrow = WpV + ((size_t)((k0 >> 1) + hf) * HDIM + n0 + ln);
    v2f b0 = wrow[0];
    v2f b1 = wrow[16];
    v2f b2 = wrow[32];
    v2f b3 = wrow[48];
    acc[0][0] = wmma4(a0, b0, acc[0][0]);
    acc[1][0] = wmma4(a1, b0, acc[1][0]);
    acc[0][1] = wmma4(a0, b1, acc[0][1]);
    acc[1][1] = wmma4(a1, b1, acc[1][1]);
    acc[0][2] = wmma4(a0, b2, acc[0][2]);
    acc[1][2] = wmma4(a1, b2, acc[1][2]);
    acc[0][3] = wmma4(a0, b3, acc[0][3]);
    acc[1][3] = wmma4(a1, b3, acc[1][3]);
  }
}

// ---------------- Kernel 1: batched input projections -----------------------
// Xout[g][T*B][H] = inputs[T*B, D] @ Wx_g[D, H] + b_g   (WpX = packed Wx)
__global__ __launch_bounds__(256)
void lstm_xproj(const float* __restrict__ X,
                const float* __restrict__ WpX,   // packed [4][256][512][2]
                const float* __restrict__ b0,  const float* __restrict__ b1,
                const float* __restrict__ b2,  const float* __restrict__ b3,
                float* __restrict__ Xout) {
  const int M  = T_STEPS * BATCH;     // 65536
  const int MT = M / 32;              // 2048 row-tiles
  const int NT = HDIM / 64;           // 8 col-strips

  int wid  = blockIdx.x * (blockDim.x >> 5) + (threadIdx.x >> 5);
  int lane = threadIdx.x & 31;
  int hf   = lane >> 4;
  int ln   = lane & 15;

  int g   = wid / (MT * NT);
  int rem = wid % (MT * NT);
  int mt  = rem / NT;
  int ntc = rem % NT;
  int m0  = mt * 32;
  int n0  = ntc * 64;

  const float* bg  = (g == 0) ? b0 : (g == 1) ? b1 : (g == 2) ? b2 : b3;
  const float* Wpg = WpX + (size_t)g * HDIM * HDIM;

  v8f acc[2][4] = {};
  gemm_strip32(X + (size_t)m0 * DIN, DIN, Wpg, n0, hf, ln, acc);

  float bb0 = bg[n0 + ln],      bb1 = bg[n0 + 16 + ln];
  float bb2 = bg[n0 + 32 + ln], bb3 = bg[n0 + 48 + ln];

  size_t base = (size_t)g * M * HDIM;
#pragma unroll
  for (int a = 0; a < 2; ++a) {
#pragma unroll
    for (int r = 0; r < 8; ++r) {
      int row = m0 + a * 16 + hf * 8 + r;     // C/D layout: M = half*8 + r
      float* orow = Xout + base + (size_t)row * HDIM + n0 + ln;
      orow[0]  = acc[a][0][r] + bb0;
      orow[16] = acc[a][1][r] + bb1;
      orow[32] = acc[a][2][r] + bb2;
      orow[48] = acc[a][3][r] + bb3;
    }
  }
}

// ---------------- Kernel 2: one recurrence step ------------------------------
struct StepParams {
  const float* Xall;       // [4][T*B][H] precomputed projections (or null)
  const float* Xt;         // inputs + t*B*D (fused path)
  const float* WpH;        // packed [4][256][512][2] recurrent weights
  const float* WpX;        // packed input weights (fused path)
  const float* bias[4];    // fused path
  const float* Hprev;      // [B, H]
  float* Hout;             // d_out + t*B*H
  float* Cst;              // [B, H] cell state (ws)
  int t;
  int fused;
};

__device__ __forceinline__ void step_elementwise(StepParams& p, const float* pre,
                                                 int tid, int nthreads, int nb0) {
  const size_t GS = (size_t)T_STEPS * BATCH * HDIM;
  for (int e = tid; e < BATCH * 64; e += nthreads) {
    int b  = e >> 6;
    int hl = e & 63;
    int h  = nb0 + hl;

    float pi = pre[(0 * 64 + b) * 64 + hl];
    float pf = pre[(1 * 64 + b) * 64 + hl];
    float po = pre[(2 * 64 + b) * 64 + hl];
    float pc = pre[(3 * 64 + b) * 64 + hl];
    if (!p.fused) {
      size_t xo = ((size_t)p.t * BATCH + b) * HDIM + h;
      pi += p.Xall[xo];
      pf += p.Xall[GS + xo];
      po += p.Xall[2 * GS + xo];
      pc += p.Xall[3 * GS + xo];
    }
    float I  = 1.0f / (1.0f + __expf(-pi));
    float F  = 1.0f / (1.0f + __expf(-pf));
    float O  = 1.0f / (1.0f + __expf(-po));
    float Ct = tanhf(pc);

    size_t so = (size_t)b * HDIM + h;
    float c = F * p.Cst[so] + I * Ct;
    p.Cst[so]  = c;
    p.Hout[so] = O * tanhf(c);
  }
}

__global__ __launch_bounds__(256)
void lstm_step(StepParams p) {
  __shared__ float pre[4 * 64 * 64];          // 64 KB: gate x batch x 64 cols

  int tid  = threadIdx.x;
  int w    = tid >> 5;                        // 0..7
  int lane = tid & 31;
  int hf   = lane >> 4;
  int ln   = lane & 15;
  int g    = w >> 1;                          // gate for this wave
  int m0   = (w & 1) * 32;                    // 32-row batch tile
  int nb0  = blockIdx.x * 64;                 // 64-wide column chunk of H

  v8f acc[2][4] = {};

  // H_{t-1} @ W_h[g]
  gemm_strip32(p.Hprev + (size_t)m0 * HDIM, HDIM,
               p.WpH + (size_t)g * HDIM * HDIM, nb0, hf, ln, acc);

  float bb0 = 0.f, bb1 = 0.f, bb2 = 0.f, bb3 = 0.f;
  if (p.fused) {
    gemm_strip32(p.Xt + (size_t)m0 * DIN, DIN,
                 p.WpX + (size_t)g * HDIM * HDIM, nb0, hf, ln, acc);
    const float* bg = p.bias[g];
    bb0 = bg[nb0 + ln];      bb1 = bg[nb0 + 16 + ln];
    bb2 = bg[nb0 + 32 + ln]; bb3 = bg[nb0 + 48 + ln];
  }

#pragma unroll
  for (int a = 0; a < 2; ++a) {
#pragma unroll
    for (int r = 0; r < 8; ++r) {
      int row = m0 + a * 16 + hf * 8 + r;
      float* prow = &pre[((size_t)g * 64 + row) * 64 + ln];
      prow[0]  = acc[a][0][r] + bb0;
      prow[16] = acc[a][1][r] + bb1;
      prow[32] = acc[a][2][r] + bb2;
      prow[48] = acc[a][3][r] + bb3;
    }
  }
  __syncthreads();
  step_elementwise(p, pre, tid, 256, nb0);
}

// ---------------- Fallback step kernel (tiny workspace, unpacked weights) ----
struct StepFbParams {
  const float* Xt;
  const float* Wh[4];
  const float* Wx[4];
  const float* bias[4];
  const float* Hprev;
  float* Hout;
  float* Cst;
};

__device__ __forceinline__ void gemm_strip16(const float* __restrict__ Arow, int K,
                                             const float* __restrict__ W, int n0,
                                             int hf, int ln, v8f (&acc)[4]) {
#pragma unroll 4
  for (int k0 = 0; k0 < K; k0 += 4) {
    v2f a = *(const v2f*)(Arow + k0 + hf * 2);
    const float* w0 = W + (size_t)(k0 + hf * 2) * HDIM + n0 + ln;
    const float* w1 = w0 + HDIM;
    v2f b;
    b.x = w0[0];  b.y = w1[0];  acc[0] = wmma4(a, b, acc[0]);
    b.x = w0[16]; b.y = w1[16]; acc[1] = wmma4(a, b, acc[1]);
    b.x = w0[32]; b.y = w1[32]; acc[2] = wmma4(a, b, acc[2]);
    b.x = w0[48]; b.y = w1[48]; acc[3] = wmma4(a, b, acc[3]);
  }
}

__global__ __launch_bounds__(512)
void lstm_step_fb(StepFbParams p) {
  __shared__ float pre[4 * 64 * 64];
  int tid  = threadIdx.x;
  int w    = tid >> 5;
  int lane = tid & 31;
  int hf   = lane >> 4;
  int ln   = lane & 15;
  int g    = w >> 2;
  int m0   = (w & 3) * 16;
  int nb0  = blockIdx.x * 64;

  v8f acc[4] = {};
  gemm_strip16(p.Hprev + (size_t)(m0 + ln) * HDIM, HDIM, p.Wh[g], nb0, hf, ln, acc);
  gemm_strip16(p.Xt + (size_t)(m0 + ln) * DIN, DIN, p.Wx[g], nb0, hf, ln, acc);
  const float* bg = p.bias[g];
  float bb0 = bg[nb0 + ln],      bb1 = bg[nb0 + 16 + ln];
  float bb2 = bg[nb0 + 32 + ln], bb3 = bg[nb0 + 48 + ln];

#pragma unroll
  for (int r = 0; r < 8; ++r) {
    int row = m0 + hf * 8 + r;
    float* prow = &pre[((size_t)g * 64 + row) * 64 + ln];
    prow[0]  = acc[0][r] + bb0;
    prow[16] = acc[1][r] + bb1;
    prow[32] = acc[2][r] + bb2;
    prow[48] = acc[3][r] + bb3;
  }
  __syncthreads();

  for (int e = tid; e < BATCH * 64; e += 512) {
    int b  = e >> 6;
    int hl = e & 63;
    int h  = nb0 + hl;
    float I  = 1.0f / (1.0f + __expf(-pre[(0 * 64 + b) * 64 + hl]));
    float F  = 1.0f / (1.0f + __expf(-pre[(1 * 64 + b) * 64 + hl]));
    float O  = 1.0f / (1.0f + __expf(-pre[(2 * 64 + b) * 64 + hl]));
    float Ct = tanhf(pre[(3 * 64 + b) * 64 + hl]);
    size_t so = (size_t)b * HDIM + h;
    float c = F * p.Cst[so] + I * Ct;
    p.Cst[so]  = c;
    p.Hout[so] = O * tanhf(c);
  }
}

// ---------------- Host driver ------------------------------------------------
extern "C" void kernel_launch(void* const* d_in, const int* in_sizes, int n_in,
                              void* d_out, int out_size, void* d_ws, size_t ws_size,
                              hipStream_t stream) {
  (void)in_sizes; (void)n_in; (void)out_size;

  const float* X   = (const float*)d_in[0];
  const float* Wxi = (const float*)d_in[1];
  const float* Whi = (const float*)d_in[2];
  const float* bi  = (const float*)d_in[3];
  const float* Wxf = (const float*)d_in[4];
  const float* Whf = (const float*)d_in[5];
  const float* bf  = (const float*)d_in[6];
  const float* Wxo = (const float*)d_in[7];
  const float* Who = (const float*)d_in[8];
  const float* bo  = (const float*)d_in[9];
  const float* Wxc = (const float*)d_in[10];
  const float* Whc = (const float*)d_in[11];
  const float* bc  = (const float*)d_in[12];
  float* out = (float*)d_out;

  const size_t GS    = (size_t)T_STEPS * BATCH * HDIM;    // one gate's projections
  const size_t STATE = (size_t)BATCH * HDIM;              // B*H
  const size_t PACK  = 8ull * HDIM * HDIM;                // packed weights (floats)

  const size_t need_small = (PACK + 2 * STATE) * sizeof(float);          // ~8.7 MB
  const size_t need_full  = need_small + 4 * GS * sizeof(float);         // ~545 MB
  const bool packed     = (ws_size >= need_small);
  const bool precompute = (ws_size >= need_full);

  float* Wp, *Cst, *H0, *Xall;
  if (packed) {
    Wp   = (float*)d_ws;          // [WpH x4][WpX x4]
    Cst  = Wp + PACK;
    H0   = Cst + STATE;
    Xall = H0 + STATE;            // only valid if precompute
  } else {
    Wp = nullptr; Xall = nullptr;
    Cst = (float*)d_ws;
    H0  = Cst + STATE;
  }

  // Zero initial H and C every call (ws is poisoned, not re-poisoned per replay)
  hipMemsetAsync(Cst, 0, STATE * sizeof(float), stream);
  hipMemsetAsync(H0,  0, STATE * sizeof(float), stream);

  if (packed) {
    PackParams pk;
    pk.src[0] = Whi; pk.src[1] = Whf; pk.src[2] = Who; pk.src[3] = Whc;
    pk.src[4] = Wxi; pk.src[5] = Wxf; pk.src[6] = Wxo; pk.src[7] = Wxc;
    pk.dst = Wp;
    pack_weights<<<dim3(4096), dim3(256), 0, stream>>>(pk);

    if (precompute) {
      const int waves = 4 * (T_STEPS * BATCH / 32) * (HDIM / 64);  // 65536
      lstm_xproj<<<dim3(waves / 8), dim3(256), 0, stream>>>(
          X, Wp + 4ull * HDIM * HDIM, bi, bf, bo, bc, Xall);
    }

    StepParams p;
    p.Xall = precompute ? Xall : nullptr;
    p.WpH  = Wp;
    p.WpX  = Wp + 4ull * HDIM * HDIM;
    p.bias[0] = bi; p.bias[1] = bf; p.bias[2] = bo; p.bias[3] = bc;
    p.Cst   = Cst;
    p.fused = precompute ? 0 : 1;
    for (int t = 0; t < T_STEPS; ++t) {
      p.t     = t;
      p.Xt    = X + (size_t)t * BATCH * DIN;
      p.Hprev = (t == 0) ? H0 : out + (size_t)(t - 1) * STATE;
      p.Hout  = out + (size_t)t * STATE;
      lstm_step<<<dim3(HDIM / 64), dim3(256), 0, stream>>>(p);
    }
  } else {
    StepFbParams p;
    p.Wh[0] = Whi; p.Wh[1] = Whf; p.Wh[2] = Who; p.Wh[3] = Whc;
    p.Wx[0] = Wxi; p.Wx[1] = Wxf; p.Wx[2] = Wxo; p.Wx[3] = Wxc;
    p.bias[0] = bi; p.bias[1] = bf; p.bias[2] = bo; p.bias[3] = bc;
    p.Cst = Cst;
    for (int t = 0; t < T_STEPS; ++t) {
      p.Xt    = X + (size_t)t * BATCH * DIN;
      p.Hprev = (t == 0) ? H0 : out + (size_t)(t - 1) * STATE;
      p.Hout  = out + (size_t)t * STATE;
      lstm_step_fb<<<dim3(HDIM / 64), dim3(512), 0, stream>>>(p);
    }
  }

  // Tail of d_out: final (H, C)
  hipMemcpyAsync(out + GS, out + (size_t)(T_STEPS - 1) * STATE,
                 STATE * sizeof(float), hipMemcpyDeviceToDevice, stream);
  hipMemcpyAsync(out + GS + STATE, Cst,
                 STATE * sizeof(float), hipMemcpyDeviceToDevice, stream);
}